// GMMflow_77292231459387
// MI455X (gfx1250) — compile-verified
//
#include <hip/hip_runtime.h>
#include <hip/hip_bf16.h>

typedef __attribute__((ext_vector_type(2))) float v2f;
typedef __attribute__((ext_vector_type(8))) float v8f;

#define B_DIM 128
#define N0_   128
#define N1_   128
#define NDIM  32
#define P_    (N0_ * N1_)      // 16384 (i,j) pairs
#define CHUNK 128              // ij columns per workgroup
#define NWGS  (P_ / CHUNK)     // 128 workgroups
#define NCOL  80               // gnum row stride: 32 (W@K) + 32 (W@H) + 1 den + pad

// ---------------------------------------------------------------------------
// Zero the global partial-accumulator (ws is poisoned by the harness).
// ---------------------------------------------------------------------------
__global__ void gmm_zero(float* __restrict__ g, int n) {
    int i = blockIdx.x * blockDim.x + threadIdx.x;
    if (i < n) g[i] = 0.0f;
}

// ---------------------------------------------------------------------------
// Fused kernel: per-(i,j) parameter tables -> GEMM1 (logw) -> exp -> GEMM2
// (weighted sums) entirely in LDS, f32 WMMA (16x16x4) for both contractions.
// Each wave (of 8) owns M-rows [16w,16w+16); the WG owns CHUNK ij columns.
// ---------------------------------------------------------------------------
__global__ __launch_bounds__(256) void gmm_main(
    const float* __restrict__ X,   // [128][32]
    const float* __restrict__ tp,  // [1]
    const float* __restrict__ Mu0, // [128][32]
    const float* __restrict__ Mu1, // [128][32]
    const float* __restrict__ S0,  // [128][32]
    const float* __restrict__ S1,  // [128][32]
    const float* __restrict__ Lam, // [128][128] flattened over ij
    float* __restrict__ gnum)      // [128][NCOL] partials (atomic)
{
    __shared__ float sB1[64][CHUNK];    // GEMM1 B: row k in {Sigma^-1 ; Mut*Sigma^-1}
    __shared__ float sB2[CHUNK][68];    // GEMM2 B: [ij][n2] n2<32:K, n2>=32:v-K*Mut (pad 68)
    __shared__ float sW[8][16][132];    // W per wave: [wave][m_local][ij_local] (pad 132)
    __shared__ float sC[CHUNK];         // per-ij logw constant

    const int tid  = threadIdx.x;
    const int lane = tid & 31;
    const int w    = tid >> 5;          // wave id 0..7
    const int h    = lane >> 4;         // half-wave
    const int nl   = lane & 15;
    const int wg   = blockIdx.x;

    const float t   = tp[0];
    const float omt = 1.0f - t;

    // ---- A1 fragments for GEMM1: rows of [-0.5*X^2 | X], K=64 (16 K-tiles of 4)
    // f32 WMMA A layout: M = lane&15, K = 4*kt + r + 2*(lane>>4)
    float a1[16][2];
    {
        const int m = 16 * w + nl;
        #pragma unroll
        for (int kt = 0; kt < 16; ++kt) {
            #pragma unroll
            for (int r = 0; r < 2; ++r) {
                int k = 4 * kt + r + 2 * h;
                float x = X[m * NDIM + (k & 31)];
                a1[kt][r] = (k < 32) ? (-0.5f * x * x) : x;
            }
        }
    }

    if (tid < CHUNK) sC[tid] = 0.0f;
    __syncthreads();

    // ---- Phase 1: per-(i,j) Gaussian-pair parameters -> LDS tables
    {
        const int ijl = tid & (CHUNK - 1);
        const int nh  = tid >> 7;                 // 0 or 1: which n half
        const int ijg = wg * CHUNK + ijl;
        const int i   = ijg >> 7;                 // N1 == 128
        const int j   = ijg & 127;
        float cpart = 0.0f;
        for (int n = nh * 16; n < nh * 16 + 16; ++n) {
            float s0  = S0[i * NDIM + n], s1 = S1[j * NDIM + n];
            float mu0 = Mu0[i * NDIM + n], mu1 = Mu1[j * NDIM + n];
            float Ds    = sqrtf(4.0f * s0 * s1 + 0.0625f);   // eps^4
            float Cs    = 0.5f * (Ds - 0.25f);               // eps^2
            float Sigma = omt * omt * s0 + t * t * s1
                        + 2.0f * t * omt * Cs + 0.25f * t * omt;
            float Pt    = t * s1 + omt * Cs;
            float Qt    = omt * s0 + t * Cs;
            float St    = Pt - Qt - 0.25f * t;
            float invS  = 1.0f / Sigma;
            float Kv    = St * invS;
            float Mut   = omt * mu0 + t * mu1;
            float vv    = mu1 - mu0;
            sB1[n][ijl]      = invS;          // pairs with -0.5*X^2
            sB1[32 + n][ijl] = Mut * invS;    // pairs with X
            sB2[ijl][n]      = Kv;            // -> multiplies X in finalize
            sB2[ijl][32 + n] = vv - Kv * Mut;
            cpart += -0.5f * (Mut * Mut * invS + logf(Sigma));
        }
        __hip_atomic_fetch_add(&sC[ijl], cpart,
                               __ATOMIC_RELAXED, __HIP_MEMORY_SCOPE_WORKGROUP);
    }
    __syncthreads();

    // ---- Phase 2: GEMM1 -> logw, then W = exp(clip(logw))*Lam into sW
    for (int ct = 0; ct < CHUNK / 16; ++ct) {
        float cval = sC[ct * 16 + nl];        // same for all M rows of the tile
        v8f acc;
        #pragma unroll
        for (int r = 0; r < 8; ++r) acc[r] = cval;
        #pragma unroll
        for (int kt = 0; kt < 16; ++kt) {
            v2f av, bv;
            av[0] = a1[kt][0];
            av[1] = a1[kt][1];
            bv[0] = sB1[4 * kt + 0 + 2 * h][ct * 16 + nl];
            bv[1] = sB1[4 * kt + 1 + 2 * h][ct * 16 + nl];
            acc = __builtin_amdgcn_wmma_f32_16x16x4_f32(
                      false, av, false, bv, (short)0, acc, false, false);
        }
        float lam = Lam[wg * CHUNK + ct * 16 + nl];
        #pragma unroll
        for (int r = 0; r < 8; ++r) {   // C layout: M = r + 8h, N = ct*16+nl
            float lw = fminf(fmaxf(acc[r], -50.0f), 50.0f);
            sW[w][r + 8 * h][ct * 16 + nl] = expf(lw) * lam;  // can be ~e^50: must stay f32
        }
    }
    __syncthreads();

    // ---- Phase 3: GEMM2: [16 x CHUNK] W-tile times [CHUNK x 80] B2 (+den col)
    v8f acc2[5];
    #pragma unroll
    for (int q = 0; q < 5; ++q)
        #pragma unroll
        for (int r = 0; r < 8; ++r) acc2[q][r] = 0.0f;

    v2f bvden;                                     // column 64 == 1.0 (den), rest 0
    bvden[0] = (nl == 0) ? 1.0f : 0.0f;
    bvden[1] = bvden[0];

    for (int kt = 0; kt < CHUNK / 4; ++kt) {
        v2f av;                                    // A: M = nl, K = 4*kt + r + 2h
        av[0] = sW[w][nl][4 * kt + 0 + 2 * h];
        av[1] = sW[w][nl][4 * kt + 1 + 2 * h];
        #pragma unroll
        for (int q = 0; q < 4; ++q) {
            v2f bv;
            bv[0] = sB2[4 * kt + 0 + 2 * h][q * 16 + nl];
            bv[1] = sB2[4 * kt + 1 + 2 * h][q * 16 + nl];
            acc2[q] = __builtin_amdgcn_wmma_f32_16x16x4_f32(
                          false, av, false, bv, (short)0, acc2[q], false, false);
        }
        acc2[4] = __builtin_amdgcn_wmma_f32_16x16x4_f32(
                      false, av, false, bvden, (short)0, acc2[4], false, false);
    }

    // ---- Atomic accumulate partials across workgroups
    #pragma unroll
    for (int q = 0; q < 4; ++q) {
        #pragma unroll
        for (int r = 0; r < 8; ++r) {
            int m = 16 * w + r + 8 * h;
            __hip_atomic_fetch_add(&gnum[m * NCOL + q * 16 + nl], acc2[q][r],
                                   __ATOMIC_RELAXED, __HIP_MEMORY_SCOPE_AGENT);
        }
    }
    if (nl == 0) {
        #pragma unroll
        for (int r = 0; r < 8; ++r) {
            int m = 16 * w + r + 8 * h;
            __hip_atomic_fetch_add(&gnum[m * NCOL + 64], acc2[4][r],
                                   __ATOMIC_RELAXED, __HIP_MEMORY_SCOPE_AGENT);
        }
    }
}

// ---------------------------------------------------------------------------
// out[b,n] = (X[b,n]*(W@K)[b,n] + (W@H)[b,n]) / den[b]
// ---------------------------------------------------------------------------
__global__ void gmm_finalize(const float* __restrict__ X,
                             const float* __restrict__ gnum,
                             float* __restrict__ out) {
    int idx = blockIdx.x * blockDim.x + threadIdx.x;
    if (idx < B_DIM * NDIM) {
        int b = idx >> 5, n = idx & 31;
        const float* g = gnum + b * NCOL;
        out[idx] = (X[idx] * g[n] + g[32 + n]) / g[64];
    }
}

extern "C" void kernel_launch(void* const* d_in, const int* in_sizes, int n_in,
                              void* d_out, int out_size, void* d_ws, size_t ws_size,
                              hipStream_t stream) {
    (void)in_sizes; (void)n_in; (void)out_size; (void)ws_size;
    const float* X   = (const float*)d_in[0];
    const float* t   = (const float*)d_in[1];
    const float* Mu0 = (const float*)d_in[2];
    const float* Mu1 = (const float*)d_in[3];
    const float* S0  = (const float*)d_in[4];
    const float* S1  = (const float*)d_in[5];
    const float* Lam = (const float*)d_in[6];
    float* gnum = (float*)d_ws;          // needs 128*80*4 = 40 KB scratch
    float* out  = (float*)d_out;

    gmm_zero<<<(B_DIM * NCOL + 255) / 256, 256, 0, stream>>>(gnum, B_DIM * NCOL);
    gmm_main<<<NWGS, 256, 0, stream>>>(X, t, Mu0, Mu1, S0, S1, Lam, gnum);
    gmm_finalize<<<(B_DIM * NDIM + 255) / 256, 256, 0, stream>>>(X, gnum, out);
}